// PointMamba_26259430048199
// MI455X (gfx1250) — compile-verified
//
#include <hip/hip_runtime.h>
#include <math.h>

// ---------------------------------------------------------------------------
// PointMamba forward for MI455X (gfx1250, wave32, WMMA).
// All GEMMs in fp32 via V_WMMA_F32_16X16X4_F32 (exact numerics; workload is
// scan-latency bound, not matmul bound, so fp32 matrix path is the right pick).
// ---------------------------------------------------------------------------

typedef float  v2f  __attribute__((ext_vector_type(2)));
typedef float  v8f  __attribute__((ext_vector_type(8)));

// ---------------------------------------------------------------------------
// Generic GEMM:  C[M,N] = act( A[M,K] * W[N,K]^T + bias )
// A row stride lda, C row stride ldc, W contiguous [N,K].
// One wave -> two 16x16 tiles stacked in M (B fragment reused by 2 WMMAs).
// 8 waves/block -> 32(M) x 128(N) block tile.
// act: 0 none, 1 relu, 2 exact gelu, 4 softplus
// GUARD=true handles K not divisible by 4 (only the tiny K=3 GEMMs).
// ---------------------------------------------------------------------------
template <bool GUARD>
__global__ __launch_bounds__(256)
void gemm_wmma(const float* __restrict__ A, const float* __restrict__ W,
               const float* __restrict__ bias, float* __restrict__ C,
               int M, int N, int K, int lda, int ldc, int act)
{
    const int wave = threadIdx.x >> 5;
    const int lane = threadIdx.x & 31;
    const int hf   = lane >> 4;      // which half of the wave (K split)
    const int mr   = lane & 15;
    const int m0   = blockIdx.y * 32;
    const int n0   = blockIdx.x * 128 + wave * 16;
    if (n0 >= N) return;

    v8f acc0 = {}, acc1 = {};
    const float* arow0 = A + (size_t)(m0 + mr) * lda;
    const float* arow1 = A + (size_t)(m0 + 16 + mr) * lda;
    const float* wrow  = W + (size_t)(n0 + mr) * K;

    if (GUARD) {
        // Rare path: ragged K (K=3). Scalar guarded loads.
        for (int k = 0; k < K; k += 4) {
            const int k0 = k + hf * 2;
            v2f a0, a1, b;
            const bool in0 = (k0 < K), in1 = (k0 + 1 < K);
            a0.x = in0 ? arow0[k0] : 0.f;  a0.y = in1 ? arow0[k0 + 1] : 0.f;
            a1.x = in0 ? arow1[k0] : 0.f;  a1.y = in1 ? arow1[k0 + 1] : 0.f;
            b.x  = in0 ? wrow[k0]  : 0.f;  b.y  = in1 ? wrow[k0 + 1]  : 0.f;
            acc0 = __builtin_amdgcn_wmma_f32_16x16x4_f32(
                false, a0, false, b, (short)0, acc0, false, false);
            acc1 = __builtin_amdgcn_wmma_f32_16x16x4_f32(
                false, a1, false, b, (short)0, acc1, false, false);
        }
    } else {
        // Fast path: K % 4 == 0, rows 8B-aligned -> unguarded b64 loads.
        // f32 WMMA 16x16x4 A fragment: VGPR0 = K0 (lanes0-15) / K2 (lanes16-31),
        // VGPR1 = K1 / K3; B 4x16 mirrors the K striping with N = lane&15.
        const v2f* a0p = (const v2f*)arow0;
        const v2f* a1p = (const v2f*)arow1;
        const v2f* wp  = (const v2f*)wrow;
        const int  kq  = K >> 2;
        #pragma unroll 4
        for (int k4 = 0; k4 < kq; ++k4) {
            v2f b  = wp [k4 * 2 + hf];
            v2f a0 = a0p[k4 * 2 + hf];
            v2f a1 = a1p[k4 * 2 + hf];
            acc0 = __builtin_amdgcn_wmma_f32_16x16x4_f32(
                false, a0, false, b, (short)0, acc0, false, false);
            acc1 = __builtin_amdgcn_wmma_f32_16x16x4_f32(
                false, a1, false, b, (short)0, acc1, false, false);
        }
    }

    // C/D layout: VGPR r -> row base+r (lanes 0-15) / base+8+r (lanes 16-31),
    // col n0 + (lane&15).
    const int   cn = n0 + mr;
    const float bv = bias ? bias[cn] : 0.f;
    #pragma unroll
    for (int t = 0; t < 2; ++t) {
        const v8f acc = t ? acc1 : acc0;
        const int cm  = m0 + t * 16 + hf * 8;
        #pragma unroll
        for (int r = 0; r < 8; ++r) {
            float v = acc[r] + bv;
            if (act == 1)      v = fmaxf(v, 0.f);
            else if (act == 2) v = 0.5f * v * (1.f + erff(v * 0.70710678118654752f));
            else if (act == 4) v = (v > 20.f) ? v : log1pf(__expf(v));
            C[(size_t)(cm + r) * ldc + cn] = v;
        }
    }
}

// ---------------------------------------------------------------------------
// Point-cloud prep
// ---------------------------------------------------------------------------
__global__ __launch_bounds__(256)
void scale_xyz_kernel(const float* __restrict__ pts, float* __restrict__ xyz,
                      float inv_scale)
{
    int idx = blockIdx.x * 256 + threadIdx.x;           // 16384*3
    if (idx >= 16384 * 3) return;
    int r = idx / 3, c = idx % 3;
    xyz[idx] = pts[r * 4 + 1 + c] * inv_scale;
}

__global__ __launch_bounds__(256)
void fps_kernel(const float* __restrict__ xyz, float* __restrict__ dist,
                int* __restrict__ cidx)
{
    const int NP = 16384, NS = 512;
    const int tid = threadIdx.x;
    __shared__ float sval[256];
    __shared__ int   sidx[256];
    __shared__ float fx, fy, fz;
    __shared__ int   far;
    for (int i = tid; i < NP; i += 256) dist[i] = 1e10f;
    if (tid == 0) far = 0;
    __syncthreads();
    for (int it = 0; it < NS; ++it) {
        if (tid == 0) {
            cidx[it] = far;
            fx = xyz[far * 3]; fy = xyz[far * 3 + 1]; fz = xyz[far * 3 + 2];
        }
        __syncthreads();
        float bv = -1.f; int bi = 0;
        for (int i = tid; i < NP; i += 256) {
            float dx = xyz[i * 3] - fx, dy = xyz[i * 3 + 1] - fy, dz = xyz[i * 3 + 2] - fz;
            float nd = fminf(dist[i], dx * dx + dy * dy + dz * dz);
            dist[i] = nd;
            if (nd > bv) { bv = nd; bi = i; }
        }
        sval[tid] = bv; sidx[tid] = bi;
        __syncthreads();
        for (int o = 128; o > 0; o >>= 1) {
            if (tid < o) {
                if (sval[tid + o] > sval[tid] ||
                    (sval[tid + o] == sval[tid] && sidx[tid + o] < sidx[tid])) {
                    sval[tid] = sval[tid + o]; sidx[tid] = sidx[tid + o];
                }
            }
            __syncthreads();
        }
        if (tid == 0) far = sidx[0];
        __syncthreads();
    }
}

__global__ __launch_bounds__(256)
void gather_centers_kernel(const float* __restrict__ xyz, const int* __restrict__ cidx,
                           float* __restrict__ center)
{
    int idx = blockIdx.x * 256 + threadIdx.x;           // 512*3
    if (idx >= 512 * 3) return;
    center[idx] = xyz[cidx[idx / 3] * 3 + (idx % 3)];
}

// One block per center: 32 rounds of lexicographic min selection.
__global__ __launch_bounds__(256)
void knn_kernel(const float* __restrict__ xyz, const float* __restrict__ center,
                int* __restrict__ knn)
{
    const int g = blockIdx.x, tid = threadIdx.x;
    __shared__ float sd[256]; __shared__ int si[256];
    __shared__ float pd; __shared__ int pi;
    const float cx = center[g * 3], cy = center[g * 3 + 1], cz = center[g * 3 + 2];
    if (tid == 0) { pd = -1.f; pi = -1; }
    __syncthreads();
    for (int r = 0; r < 32; ++r) {
        const float prevD = pd; const int prevI = pi;
        float bv = 3.4e38f; int bi = 1 << 30;
        for (int i = tid; i < 16384; i += 256) {
            float dx = xyz[i * 3] - cx, dy = xyz[i * 3 + 1] - cy, dz = xyz[i * 3 + 2] - cz;
            float d = dx * dx + dy * dy + dz * dz;
            bool gtprev = (d > prevD) || (d == prevD && i > prevI);
            if (gtprev && (d < bv || (d == bv && i < bi))) { bv = d; bi = i; }
        }
        sd[tid] = bv; si[tid] = bi;
        __syncthreads();
        for (int o = 128; o > 0; o >>= 1) {
            if (tid < o) {
                if (sd[tid + o] < sd[tid] ||
                    (sd[tid + o] == sd[tid] && si[tid + o] < si[tid])) {
                    sd[tid] = sd[tid + o]; si[tid] = si[tid + o];
                }
            }
            __syncthreads();
        }
        if (tid == 0) { knn[g * 32 + r] = si[0]; pd = sd[0]; pi = si[0]; }
        __syncthreads();
    }
}

__global__ __launch_bounds__(256)
void neigh_kernel(const float* __restrict__ xyz, const int* __restrict__ knn,
                  const float* __restrict__ center, float* __restrict__ neigh)
{
    int idx = blockIdx.x * 256 + threadIdx.x;           // 16384*3
    if (idx >= 16384 * 3) return;
    int row = idx / 3, c = idx % 3;
    int g = row / 32;
    int j = knn[row];
    neigh[idx] = xyz[j * 3 + c] - center[g * 3 + c];
}

// ---------------------------------------------------------------------------
// BatchNorm helpers (stats over all M rows per channel)
// ---------------------------------------------------------------------------
__global__ __launch_bounds__(256)
void bn_stats_kernel(const float* __restrict__ X, int M, int C,
                     float* __restrict__ mean, float* __restrict__ var)
{
    const int c = blockIdx.x, tid = threadIdx.x;
    float s = 0.f, s2 = 0.f;
    for (int m = tid; m < M; m += 256) {
        float v = X[(size_t)m * C + c];
        s += v; s2 += v * v;
    }
    __shared__ float rs[256], rs2[256];
    rs[tid] = s; rs2[tid] = s2;
    __syncthreads();
    for (int o = 128; o > 0; o >>= 1) {
        if (tid < o) { rs[tid] += rs[tid + o]; rs2[tid] += rs2[tid + o]; }
        __syncthreads();
    }
    if (tid == 0) {
        float mu = rs[0] / M;
        mean[c] = mu;
        var[c]  = rs2[0] / M - mu * mu;
    }
}

__global__ __launch_bounds__(256)
void bn_apply_relu_kernel(float* __restrict__ X, const float* __restrict__ mean,
                          const float* __restrict__ var, const float* __restrict__ g,
                          const float* __restrict__ b, int total, int C)
{
    int idx = blockIdx.x * 256 + threadIdx.x;
    if (idx >= total) return;
    int c = idx % C;
    float v = (X[idx] - mean[c]) * rsqrtf(var[c] + 1e-5f) * g[c] + b[c];
    X[idx] = fmaxf(v, 0.f);
}

__global__ __launch_bounds__(256)
void group_max_kernel(const float* __restrict__ F, float* __restrict__ FG, int C)
{
    int idx = blockIdx.x * 256 + threadIdx.x;           // 512*C
    if (idx >= 512 * C) return;
    int g = idx / C, c = idx % C;
    float m = -3.4e38f;
    for (int n = 0; n < 32; ++n)
        m = fmaxf(m, F[((size_t)g * 32 + n) * C + c]);
    FG[(size_t)g * C + c] = m;
}

__global__ __launch_bounds__(256)
void concat_kernel(const float* __restrict__ F2, const float* __restrict__ FG,
                   float* __restrict__ cat)
{
    int idx = blockIdx.x * 256 + threadIdx.x;           // 16384*512
    if (idx >= 16384 * 512) return;
    int row = idx / 512, c = idx % 512;
    int g = row / 32;
    cat[idx] = (c < 256) ? FG[(size_t)g * 256 + c]
                         : F2[(size_t)row * 256 + (c - 256)];
}

// Stable ascending argsort of center[:,a] via rank counting (3 axes).
__global__ __launch_bounds__(512)
void argsort3_kernel(const float* __restrict__ center, int* __restrict__ ord)
{
    const int a = blockIdx.x, i = threadIdx.x;          // blockDim 512
    __shared__ float key[512];
    key[i] = center[i * 3 + a];
    __syncthreads();
    const float ki = key[i];
    int r = 0;
    for (int j = 0; j < 512; ++j) {
        float kj = key[j];
        r += (kj < ki) || (kj == ki && j < i);
    }
    ord[a * 512 + r] = i;
}

__global__ __launch_bounds__(256)
void build_res_kernel(const float* __restrict__ tok, const float* __restrict__ pos,
                      const int* __restrict__ ord, float* __restrict__ res)
{
    int idx = blockIdx.x * 256 + threadIdx.x;           // 1536*512
    if (idx >= 1536 * 512) return;
    int l = idx / 512, c = idx % 512;
    int g = ord[l];
    res[idx] = tok[(size_t)g * 512 + c] + pos[(size_t)g * 512 + c];
}

// ---------------------------------------------------------------------------
// Mamba pieces
// ---------------------------------------------------------------------------
__global__ __launch_bounds__(256)
void ln_kernel(const float* __restrict__ X, float* __restrict__ Y,
               const float* __restrict__ w, const float* __restrict__ b, int C)
{
    const int row = blockIdx.x, tid = threadIdx.x;
    const float* x = X + (size_t)row * C;
    float s = 0.f, s2 = 0.f;
    for (int c = tid; c < C; c += 256) { float v = x[c]; s += v; s2 += v * v; }
    __shared__ float rs[256], rs2[256];
    rs[tid] = s; rs2[tid] = s2;
    __syncthreads();
    for (int o = 128; o > 0; o >>= 1) {
        if (tid < o) { rs[tid] += rs[tid + o]; rs2[tid] += rs2[tid + o]; }
        __syncthreads();
    }
    float mu  = rs[0] / C;
    float rin = rsqrtf(rs2[0] / C - mu * mu + 1e-5f);
    for (int c = tid; c < C; c += 256)
        Y[(size_t)row * C + c] = (x[c] - mu) * rin * w[c] + b[c];
}

__global__ __launch_bounds__(256)
void residual_add_kernel(float* __restrict__ res, const float* __restrict__ h, int n)
{
    int idx = blockIdx.x * 256 + threadIdx.x;
    if (idx < n) res[idx] += h[idx];
}

// depthwise causal conv (D_CONV=4) over first 1024 cols of xz, then SiLU
__global__ __launch_bounds__(256)
void conv_silu_kernel(const float* __restrict__ xz, const float* __restrict__ cw,
                      const float* __restrict__ cb, float* __restrict__ out)
{
    int idx = blockIdx.x * 256 + threadIdx.x;           // 1536*1024
    if (idx >= 1536 * 1024) return;
    int l = idx >> 10, d = idx & 1023;
    float acc = cb[d];
    #pragma unroll
    for (int k = 0; k < 4; ++k) {
        int ls = l - 3 + k;
        if (ls >= 0) acc += xz[(size_t)ls * 2048 + d] * cw[d * 4 + k];
    }
    out[idx] = acc / (1.f + __expf(-acc));
}

// Sequential selective scan. 1024 threads = one channel each. B/C for the
// whole sequence staged through LDS in 48KB chunks (CDNA5: 320KB LDS/WGP).
__global__ __launch_bounds__(1024)
void scan_kernel(const float* __restrict__ xc, const float* __restrict__ dt,
                 const float* __restrict__ xdbl, const float* __restrict__ Alog,
                 float* __restrict__ y)
{
    const int CH = 1024;
    const int d = threadIdx.x;
    __shared__ float sBC[384 * 32];                     // 48KB chunk of (B|C)
    float a[16], h[16];
    #pragma unroll
    for (int n = 0; n < 16; ++n) {
        a[n] = -__expf(Alog[d * 16 + n]);
        h[n] = 0.f;
    }
    for (int c = 0; c < 4; ++c) {
        __syncthreads();
        const int l0 = c * 384;
        for (int idx = d; idx < 384 * 32; idx += 1024) {
            int ll = idx >> 5, n = idx & 31;
            sBC[idx] = xdbl[(size_t)(l0 + ll) * 64 + 32 + n];   // B then C
        }
        __syncthreads();
        for (int ll = 0; ll < 384; ++ll) {
            const int l = l0 + ll;
            const float dtv = dt[(size_t)l * CH + d];
            const float xv  = xc[(size_t)l * CH + d];
            const float dx  = dtv * xv;
            const float* bc = &sBC[ll * 32];
            float acc = 0.f;
            #pragma unroll
            for (int n = 0; n < 16; ++n) {
                h[n] = __expf(dtv * a[n]) * h[n] + dx * bc[n];
                acc += h[n] * bc[16 + n];
            }
            y[(size_t)l * CH + d] = acc;
        }
    }
}

// y = (y + xc*D) * silu(z), z = xz[:,1024:2048]
__global__ __launch_bounds__(256)
void gate_kernel(float* __restrict__ y, const float* __restrict__ xc,
                 const float* __restrict__ xz, const float* __restrict__ Dp)
{
    int idx = blockIdx.x * 256 + threadIdx.x;           // 1536*1024
    if (idx >= 1536 * 1024) return;
    int l = idx >> 10, d = idx & 1023;
    float z = xz[(size_t)l * 2048 + 1024 + d];
    float v = y[idx] + xc[idx] * Dp[d];
    y[idx] = v * (z / (1.f + __expf(-z)));
}

__global__ __launch_bounds__(256)
void cent_kernel(const float* __restrict__ center, const int* __restrict__ limit,
                 float* __restrict__ out)
{
    int idx = blockIdx.x * 256 + threadIdx.x;           // 1536*4
    if (idx >= 1536 * 4) return;
    int r = idx >> 2, c = idx & 3;
    float v = 0.f;
    if (c != 0) {
        float lim = (float)(limit[0] - 1);
        v = center[(r / 3) * 3 + (c - 1)];
        v = fminf(fmaxf(v, 0.f), lim);
    }
    out[idx] = v;
}

// ---------------------------------------------------------------------------
// Host orchestration
// ---------------------------------------------------------------------------
extern "C" void kernel_launch(void* const* d_in, const int* in_sizes, int n_in,
                              void* d_out, int out_size, void* d_ws, size_t ws_size,
                              hipStream_t stream)
{
    (void)in_sizes; (void)n_in; (void)out_size; (void)ws_size;

    const float* pts      = (const float*)d_in[0];
    const int*   slim     = (const int*)  d_in[1];
    const float* enc_c1w  = (const float*)d_in[2];
    const float* enc_c1b  = (const float*)d_in[3];
    const float* enc_bn1g = (const float*)d_in[4];
    const float* enc_bn1b = (const float*)d_in[5];
    const float* enc_c2w  = (const float*)d_in[6];
    const float* enc_c2b  = (const float*)d_in[7];
    const float* enc_c3w  = (const float*)d_in[8];
    const float* enc_c3b  = (const float*)d_in[9];
    const float* enc_bn2g = (const float*)d_in[10];
    const float* enc_bn2b = (const float*)d_in[11];
    const float* enc_c4w  = (const float*)d_in[12];
    const float* enc_c4b  = (const float*)d_in[13];
    const float* pos_w1   = (const float*)d_in[14];
    const float* pos_b1   = (const float*)d_in[15];
    const float* pos_w2   = (const float*)d_in[16];
    const float* pos_b2   = (const float*)d_in[17];
    const float* ln_w     = (const float*)d_in[18];
    const float* ln_b     = (const float*)d_in[19];
    const float* in_proj  = (const float*)d_in[20];
    const float* conv_w   = (const float*)d_in[21];
    const float* conv_b   = (const float*)d_in[22];
    const float* x_proj   = (const float*)d_in[23];
    const float* dt_w     = (const float*)d_in[24];
    const float* dt_b     = (const float*)d_in[25];
    const float* A_log    = (const float*)d_in[26];
    const float* Dp       = (const float*)d_in[27];
    const float* out_proj = (const float*)d_in[28];
    const float* normf_w  = (const float*)d_in[29];
    const float* normf_b  = (const float*)d_in[30];
    const float* norm_w   = (const float*)d_in[31];
    const float* norm_b   = (const float*)d_in[32];
    const float* adj_w    = (const float*)d_in[33];
    const float* adj_b    = (const float*)d_in[34];

    float* Wf = (float*)d_ws;
    // two 32MB arenas + small region
    float* P0 = Wf;                    // 8,388,608 floats
    float* P2 = Wf + 8388608;          // 8,388,608 floats
    float* S  = Wf + 16777216;
    float* xyz    = S;                 // 49152
    float* dist   = S + 49152;         // 16384
    float* neigh  = S + 65536;         // 49152
    float* center = S + 114688;        // 1536
    float* fg     = S + 116224;        // 131072
    float* tok    = S + 247296;        // 262144
    float* posb   = S + 509440;        // 262144
    float* posm   = S + 771584;        // 131072
    float* bnm    = S + 902656;        // 512
    float* bnv    = S + 903168;        // 512
    float* res    = S + 903680;        // 786432
    float* hbuf   = S + 1690112;       // 786432
    int*   iS     = (int*)(S + 2476544);
    int*   cidx   = iS;                // 512
    int*   knn    = iS + 512;          // 16384
    int*   ord    = iS + 16896;        // 1536

    // encoder aliases inside arenas
    float* f1  = P0;                   // 16384x128
    float* f2  = P0 + 2097152;         // 16384x256
    float* cat = P2;                   // 16384x512
    float* f3  = P0;                   // 16384x512 (f1/f2 consumed)
    float* f4  = P2;                   // 16384x512 (cat consumed)
    // mamba aliases inside arenas
    float* xz   = P0;                  // 1536x2048
    float* xcv  = P0 + 3145728;        // 1536x1024
    float* dtb  = P0 + 4718592;        // 1536x1024
    float* ybuf = P0 + 6291456;        // 1536x1024
    float* xdbl = P0 + 7864320;        // 1536x64
    float* hn   = P2;                  // 1536x512

    float* out_tok  = (float*)d_out;                  // 4*1536*512
    float* out_cent = (float*)d_out + 4 * 1536 * 512; // 4*1536*4

    auto gemm = [&](const float* A, const float* Wt, const float* bias, float* Cp,
                    int M, int N, int K, int lda, int ldc, int act) {
        dim3 grid((N + 127) / 128, M / 32);
        if ((K % 4) == 0 && (lda % 2) == 0)
            gemm_wmma<false><<<grid, 256, 0, stream>>>(A, Wt, bias, Cp, M, N, K, lda, ldc, act);
        else
            gemm_wmma<true><<<grid, 256, 0, stream>>>(A, Wt, bias, Cp, M, N, K, lda, ldc, act);
    };
    auto blocks = [](int n) { return (n + 255) / 256; };

    const float scales[4] = {2.f, 4.f, 6.f, 8.f};
    for (int s = 0; s < 4; ++s) {
        // ---- grouping ----
        scale_xyz_kernel<<<blocks(16384 * 3), 256, 0, stream>>>(pts, xyz, 1.f / scales[s]);
        fps_kernel<<<1, 256, 0, stream>>>(xyz, dist, cidx);
        gather_centers_kernel<<<blocks(512 * 3), 256, 0, stream>>>(xyz, cidx, center);
        knn_kernel<<<512, 256, 0, stream>>>(xyz, center, knn);
        neigh_kernel<<<blocks(16384 * 3), 256, 0, stream>>>(xyz, knn, center, neigh);

        // ---- encoder ----
        gemm(neigh, enc_c1w, enc_c1b, f1, 16384, 128, 3, 3, 128, 0);
        bn_stats_kernel<<<128, 256, 0, stream>>>(f1, 16384, 128, bnm, bnv);
        bn_apply_relu_kernel<<<blocks(16384 * 128), 256, 0, stream>>>(
            f1, bnm, bnv, enc_bn1g, enc_bn1b, 16384 * 128, 128);
        gemm(f1, enc_c2w, enc_c2b, f2, 16384, 256, 128, 128, 256, 0);
        group_max_kernel<<<blocks(512 * 256), 256, 0, stream>>>(f2, fg, 256);
        concat_kernel<<<blocks(16384 * 512), 256, 0, stream>>>(f2, fg, cat);
        gemm(cat, enc_c3w, enc_c3b, f3, 16384, 512, 512, 512, 512, 0);
        bn_stats_kernel<<<512, 256, 0, stream>>>(f3, 16384, 512, bnm, bnv);
        bn_apply_relu_kernel<<<blocks(16384 * 512), 256, 0, stream>>>(
            f3, bnm, bnv, enc_bn2g, enc_bn2b, 16384 * 512, 512);
        gemm(f3, enc_c4w, enc_c4b, f4, 16384, 512, 512, 512, 512, 0);
        group_max_kernel<<<blocks(512 * 512), 256, 0, stream>>>(f4, tok, 512);

        // ---- positional MLP ----
        gemm(center, pos_w1, pos_b1, posm, 512, 256, 3, 3, 256, 2);
        gemm(posm, pos_w2, pos_b2, posb, 512, 512, 256, 256, 512, 0);

        // ---- serialize tokens by 3 axis orders ----
        argsort3_kernel<<<3, 512, 0, stream>>>(center, ord);
        build_res_kernel<<<blocks(1536 * 512), 256, 0, stream>>>(tok, posb, ord, res);

        // ---- 12 Mamba layers ----
        for (int i = 0; i < 12; ++i) {
            if (i > 0)
                residual_add_kernel<<<blocks(1536 * 512), 256, 0, stream>>>(res, hbuf, 1536 * 512);
            ln_kernel<<<1536, 256, 0, stream>>>(res, hn, ln_w + i * 512, ln_b + i * 512, 512);
            gemm(hn, in_proj + (size_t)i * 2048 * 512, nullptr, xz,
                 1536, 2048, 512, 512, 2048, 0);
            conv_silu_kernel<<<blocks(1536 * 1024), 256, 0, stream>>>(
                xz, conv_w + (size_t)i * 1024 * 4, conv_b + (size_t)i * 1024, xcv);
            gemm(xcv, x_proj + (size_t)i * 64 * 1024, nullptr, xdbl,
                 1536, 64, 1024, 1024, 64, 0);
            gemm(xdbl, dt_w + (size_t)i * 1024 * 32, dt_b + (size_t)i * 1024, dtb,
                 1536, 1024, 32, 64, 1024, 4 /*softplus*/);
            scan_kernel<<<1, 1024, 0, stream>>>(xcv, dtb, xdbl,
                                                A_log + (size_t)i * 1024 * 16, ybuf);
            gate_kernel<<<blocks(1536 * 1024), 256, 0, stream>>>(
                ybuf, xcv, xz, Dp + (size_t)i * 1024);
            gemm(ybuf, out_proj + (size_t)i * 512 * 1024, nullptr, hbuf,
                 1536, 512, 1024, 1024, 512, 0);
        }
        residual_add_kernel<<<blocks(1536 * 512), 256, 0, stream>>>(res, hbuf, 1536 * 512);

        // ---- final norms + projection ----
        ln_kernel<<<1536, 256, 0, stream>>>(res, hn, normf_w, normf_b, 512);
        ln_kernel<<<1536, 256, 0, stream>>>(hn, res, norm_w, norm_b, 512);
        gemm(res, adj_w, adj_b, out_tok + (size_t)s * 1536 * 512,
             1536, 512, 512, 512, 512, 0);

        // ---- centers output ----
        cent_kernel<<<blocks(1536 * 4), 256, 0, stream>>>(
            center, slim, out_cent + (size_t)s * 1536 * 4);
    }
}